// SGCN_34342558499196
// MI455X (gfx1250) — compile-verified
//
#include <hip/hip_runtime.h>
#include <math.h>

typedef __attribute__((ext_vector_type(2))) float v2f;
typedef __attribute__((ext_vector_type(8))) float v8f;

// ---------------------------------------------------------------------------
// zero fill
// ---------------------------------------------------------------------------
__global__ void zero_f32(float* __restrict__ p, long long count) {
  long long t = (long long)blockIdx.x * blockDim.x + threadIdx.x;
  if (t < count) p[t] = 0.0f;
}

// ---------------------------------------------------------------------------
// Layer-1 scatter: sum[dst][0:64] += x[src][0:64]; cnt[dst] += 1
// 16 threads / edge, each does a float4 worth of atomics (L2-resident target)
// ---------------------------------------------------------------------------
__global__ void scatter_x64(const long long* __restrict__ ei,
                            const float* __restrict__ x,
                            float* __restrict__ sum, float* __restrict__ cnt,
                            long long E) {
  long long t = (long long)blockIdx.x * blockDim.x + threadIdx.x;
  if (t >= E * 16) return;
  long long e = t >> 4;
  int q = (int)(t & 15);
  long long s = ei[e];
  long long d = ei[E + e];
  const float4 v = *(const float4*)(x + s * 64 + q * 4);
  float* o = sum + d * 64 + q * 4;
  atomicAdd(o + 0, v.x);
  atomicAdd(o + 1, v.y);
  atomicAdd(o + 2, v.z);
  atomicAdd(o + 3, v.w);
  if (q == 0) atomicAdd(cnt + d, 1.0f);
}

// ---------------------------------------------------------------------------
// Layer-2 scatter: sum[dst][0:32] += fA[src], sum[dst][32:64] += fB[src]
// ---------------------------------------------------------------------------
__global__ void scatter_z32x2(const long long* __restrict__ ei,
                              const float* __restrict__ fA,
                              const float* __restrict__ fB,
                              float* __restrict__ sum, long long E) {
  long long t = (long long)blockIdx.x * blockDim.x + threadIdx.x;
  if (t >= E * 16) return;
  long long e = t >> 4;
  int q = (int)(t & 15);
  long long s = ei[e];
  long long d = ei[E + e];
  const float* f = (q < 8) ? fA : fB;
  int c = (q & 7) * 4;
  const float4 v = *(const float4*)(f + s * 32 + c);
  float* o = sum + d * 64 + ((q < 8) ? 0 : 32) + c;
  atomicAdd(o + 0, v.x);
  atomicAdd(o + 1, v.y);
  atomicAdd(o + 2, v.z);
  atomicAdd(o + 3, v.w);
}

// ---------------------------------------------------------------------------
// Layer 1: per 16-node tile, A = [mean(64) | x(64)] (K=128) staged in LDS,
// D = tanh(A @ W + b) via V_WMMA_F32_16X16X4_F32, two 16-col output tiles.
// One wave per tile, 4 waves / block.
// ---------------------------------------------------------------------------
__global__ __launch_bounds__(128) void sgcn_layer1(
    const float* __restrict__ x,
    const float* __restrict__ sum_p, const float* __restrict__ cnt_p,
    const float* __restrict__ sum_n, const float* __restrict__ cnt_n,
    const float* __restrict__ W1p, const float* __restrict__ b1p,
    const float* __restrict__ W1n, const float* __restrict__ b1n,
    float* __restrict__ zp, float* __restrict__ zn, int n) {
  __shared__ float As[4][16][132];  // padded to dodge bank conflicts
  const int lane = threadIdx.x & 31;
  const int wave = threadIdx.x >> 5;
  const int tileBase = blockIdx.x * 64 + wave * 16;

  const int rA = lane & 15;            // A row for this lane
  const int kSel = (lane >> 4) << 1;   // lanes 16-31 carry K+2/K+3
  const int cB = lane & 15;            // B/D column within 16-wide tile
  const int rowOff = (lane >> 4) << 3; // lanes 16-31 hold rows M+8

  for (int pass = 0; pass < 2; ++pass) {
    const float* sum = pass ? sum_n : sum_p;
    const float* cnt = pass ? cnt_n : cnt_p;
    const float* W = pass ? W1n : W1p;
    const float* b = pass ? b1n : b1p;
    float* zout = pass ? zn : zp;

    __syncthreads();
    // stage A tile: cols 0-63 = segment mean, cols 64-127 = x
    for (int i = lane; i < 16 * 128; i += 32) {
      int row = i >> 7;
      int col = i & 127;
      int g = tileBase + row;
      float v = 0.0f;
      if (g < n) {
        if (col < 64) {
          float invc = 1.0f / fmaxf(cnt[g], 1.0f);
          v = sum[(long long)g * 64 + col] * invc;
        } else {
          v = x[(long long)g * 64 + (col - 64)];
        }
      }
      As[wave][row][col] = v;
    }
    __syncthreads();

    v8f c0 = {}, c1 = {};
    for (int k0 = 0; k0 < 128; k0 += 4) {
      int kb = k0 + kSel;
      v2f a, b0, b1;
      a.x = As[wave][rA][kb];
      a.y = As[wave][rA][kb + 1];
      b0.x = W[kb * 32 + cB];
      b0.y = W[(kb + 1) * 32 + cB];
      b1.x = W[kb * 32 + 16 + cB];
      b1.y = W[(kb + 1) * 32 + 16 + cB];
      c0 = __builtin_amdgcn_wmma_f32_16x16x4_f32(false, a, false, b0, (short)0,
                                                 c0, false, false);
      c1 = __builtin_amdgcn_wmma_f32_16x16x4_f32(false, a, false, b1, (short)0,
                                                 c1, false, false);
    }

    float bc0 = b[cB], bc1 = b[16 + cB];
#pragma unroll
    for (int r = 0; r < 8; ++r) {
      int row = tileBase + r + rowOff;
      if (row < n) {
        zout[(long long)row * 32 + cB] = tanhf(c0[r] + bc0);
        zout[(long long)row * 32 + 16 + cB] = tanhf(c1[r] + bc1);
      }
    }
  }
}

// ---------------------------------------------------------------------------
// Layer 2: A = [pos-mean | neg-mean | z_self] (K=96); output -> d_out (N x 64).
// pos pass -> cols 0-31, neg pass -> cols 32-63.
// ---------------------------------------------------------------------------
__global__ __launch_bounds__(128) void sgcn_layer2(
    const float* __restrict__ zp, const float* __restrict__ zn,
    const float* __restrict__ sum2p, const float* __restrict__ cnt_p,
    const float* __restrict__ sum2n, const float* __restrict__ cnt_n,
    const float* __restrict__ W2p, const float* __restrict__ b2p,
    const float* __restrict__ W2n, const float* __restrict__ b2n,
    float* __restrict__ out, int n) {
  __shared__ float As[4][16][100];
  const int lane = threadIdx.x & 31;
  const int wave = threadIdx.x >> 5;
  const int tileBase = blockIdx.x * 64 + wave * 16;

  const int rA = lane & 15;
  const int kSel = (lane >> 4) << 1;
  const int cB = lane & 15;
  const int rowOff = (lane >> 4) << 3;

  for (int pass = 0; pass < 2; ++pass) {
    const float* W = pass ? W2n : W2p;
    const float* b = pass ? b2n : b2p;
    const float* zself = pass ? zn : zp;
    const int halfOff = pass ? 32 : 0;  // which 32-wide half of the sums

    __syncthreads();
    for (int i = lane; i < 16 * 96; i += 32) {
      int row = i / 96;
      int col = i % 96;
      int g = tileBase + row;
      float v = 0.0f;
      if (g < n) {
        if (col < 32) {
          float invc = 1.0f / fmaxf(cnt_p[g], 1.0f);
          v = sum2p[(long long)g * 64 + halfOff + col] * invc;
        } else if (col < 64) {
          float invc = 1.0f / fmaxf(cnt_n[g], 1.0f);
          v = sum2n[(long long)g * 64 + halfOff + (col - 32)] * invc;
        } else {
          v = zself[(long long)g * 32 + (col - 64)];
        }
      }
      As[wave][row][col] = v;
    }
    __syncthreads();

    v8f c0 = {}, c1 = {};
    for (int k0 = 0; k0 < 96; k0 += 4) {
      int kb = k0 + kSel;
      v2f a, b0, b1;
      a.x = As[wave][rA][kb];
      a.y = As[wave][rA][kb + 1];
      b0.x = W[kb * 32 + cB];
      b0.y = W[(kb + 1) * 32 + cB];
      b1.x = W[kb * 32 + 16 + cB];
      b1.y = W[(kb + 1) * 32 + 16 + cB];
      c0 = __builtin_amdgcn_wmma_f32_16x16x4_f32(false, a, false, b0, (short)0,
                                                 c0, false, false);
      c1 = __builtin_amdgcn_wmma_f32_16x16x4_f32(false, a, false, b1, (short)0,
                                                 c1, false, false);
    }

    float bc0 = b[cB], bc1 = b[16 + cB];
#pragma unroll
    for (int r = 0; r < 8; ++r) {
      int row = tileBase + r + rowOff;
      if (row < n) {
        out[(long long)row * 64 + pass * 32 + cB] = tanhf(c0[r] + bc0);
        out[(long long)row * 64 + pass * 32 + 16 + cB] = tanhf(c1[r] + bc1);
      }
    }
  }
}

// ---------------------------------------------------------------------------
extern "C" void kernel_launch(void* const* d_in, const int* in_sizes, int n_in,
                              void* d_out, int out_size, void* d_ws,
                              size_t ws_size, hipStream_t stream) {
  const float* x = (const float*)d_in[0];
  const float* W1p = (const float*)d_in[1];
  const float* b1p = (const float*)d_in[2];
  const float* W1n = (const float*)d_in[3];
  const float* b1n = (const float*)d_in[4];
  const float* W2p = (const float*)d_in[5];
  const float* b2p = (const float*)d_in[6];
  const float* W2n = (const float*)d_in[7];
  const float* b2n = (const float*)d_in[8];
  const long long* pe = (const long long*)d_in[9];
  const long long* ne = (const long long*)d_in[10];

  const int n = in_sizes[0] / 64;                  // 100000 nodes
  const long long E = (long long)in_sizes[9] / 2;  // 1.25M edges / type

  // workspace carve-out (floats): [sum_p n*64][sum_n n*64][cnt_p n][cnt_n n][zp n*32][zn n*32]
  float* ws = (float*)d_ws;
  float* sum_p = ws;
  float* sum_n = sum_p + (long long)n * 64;
  float* cnt_p = sum_n + (long long)n * 64;
  float* cnt_n = cnt_p + n;
  float* zp = cnt_n + n;
  float* zn = zp + (long long)n * 32;

  const int TB = 256;
  const long long st = E * 16;
  const int sblocks = (int)((st + TB - 1) / TB);
  const int lblocks = (n + 63) / 64;

  // ---- layer 1 ----
  long long zc1 = (long long)n * 128 + 2LL * n;  // sums + counts
  zero_f32<<<(int)((zc1 + TB - 1) / TB), TB, 0, stream>>>(sum_p, zc1);
  scatter_x64<<<sblocks, TB, 0, stream>>>(pe, x, sum_p, cnt_p, E);
  scatter_x64<<<sblocks, TB, 0, stream>>>(ne, x, sum_n, cnt_n, E);
  sgcn_layer1<<<lblocks, 128, 0, stream>>>(x, sum_p, cnt_p, sum_n, cnt_n, W1p,
                                           b1p, W1n, b1n, zp, zn, n);

  // ---- layer 2 (counts are edge-only: reuse; re-zero sums only) ----
  long long zc2 = (long long)n * 128;
  zero_f32<<<(int)((zc2 + TB - 1) / TB), TB, 0, stream>>>(sum_p, zc2);
  // pos edges: [0:32]+=zp (feeds out_p blk0), [32:64]+=zn (feeds out_n blk0)
  scatter_z32x2<<<sblocks, TB, 0, stream>>>(pe, zp, zn, sum_p, E);
  // neg edges: [0:32]+=zn (feeds out_p blk1), [32:64]+=zp (feeds out_n blk1)
  scatter_z32x2<<<sblocks, TB, 0, stream>>>(ne, zn, zp, sum_n, E);
  sgcn_layer2<<<lblocks, 128, 0, stream>>>(zp, zn, sum_p, cnt_p, sum_n, cnt_n,
                                           W2p, b2p, W2n, b2n, (float*)d_out,
                                           n);
}